// GCN_5403068858882
// MI455X (gfx1250) — compile-verified
//
#include <hip/hip_runtime.h>
#include <cmath>

#define H 64
#define NLAYERS 3

typedef __attribute__((ext_vector_type(2))) float v2f;
typedef __attribute__((ext_vector_type(8))) float v8f;

// ---- monotonic float <-> ordered-uint keys (key 0 < key of every real float) ----
__device__ __forceinline__ unsigned fkey(float x) {
    unsigned b = __float_as_uint(x);
    return (b & 0x80000000u) ? ~b : (b | 0x80000000u);
}
__device__ __forceinline__ float funkey(unsigned k) {
    return (k & 0x80000000u) ? __uint_as_float(k & 0x7fffffffu) : __uint_as_float(~k);
}

// =====================================================================
// Generic f32 WMMA GEMM: out = residual + maybe_relu(((A@W)+bias_pre)*scale*scale_const + bias)
// One wave (32 threads) computes a 16x16 tile via V_WMMA_F32_16X16X4_F32.
// =====================================================================
__global__ __launch_bounds__(32) void k_wmma_gemm(
    const float* __restrict__ A, int lda,
    const float* __restrict__ W, int ldw,
    const float* __restrict__ bias_pre,
    const float* __restrict__ scale, float scale_const,
    const float* __restrict__ bias,
    const float* __restrict__ residual, int ldr,
    float* __restrict__ out, int ldc,
    int M, int K, int do_relu)
{
    const int lane = threadIdx.x;
    const int hl   = lane >> 4;     // 0 or 1
    const int l    = lane & 15;
    int rowA = blockIdx.x * 16 + l;
    if (rowA > M - 1) rowA = M - 1;                 // clamp (no divergence before WMMA)
    const int colB = blockIdx.y * 16 + l;
    const float* Ar = A + (size_t)rowA * lda;

    v8f acc = {};
    for (int k0 = 0; k0 < K; k0 += 4) {
        const int ka = k0 + 2 * hl;                 // ISA A layout: lanes16-31 hold K+2,K+3
        v2f a, b;
        a.x = Ar[ka];
        a.y = Ar[ka + 1];
        b.x = W[(size_t)ka * ldw + colB];           // ISA B layout: lanes16-31 hold rows K+2,K+3
        b.y = W[(size_t)(ka + 1) * ldw + colB];
        acc = __builtin_amdgcn_wmma_f32_16x16x4_f32(
            /*neg_a=*/false, a, /*neg_b=*/false, b,
            /*c_mod=*/(short)0, acc, /*reuse_a=*/false, /*reuse_b=*/false);
    }

    const int row0 = blockIdx.x * 16 + hl * 8;      // C/D layout: VGPR r -> M = r + 8*half
    #pragma unroll
    for (int r = 0; r < 8; ++r) {
        const int row = row0 + r;
        if (row >= M) continue;
        float v = acc[r];
        if (bias_pre) v += bias_pre[colB];
        if (scale)    v *= scale[colB];
        v *= scale_const;
        if (bias)     v += bias[colB];
        if (do_relu)  v = fmaxf(v, 0.0f);
        if (residual) v += residual[(size_t)row * ldr + colB];
        out[(size_t)row * ldc + colB] = v;
    }
}

// =====================================================================
// GCNConv stage
// =====================================================================
__global__ void k_h0pre(const float* __restrict__ x, const float* __restrict__ w,
                        float* __restrict__ h0, int N)
{
    int tid = blockIdx.x * blockDim.x + threadIdx.x;
    if (tid >= N * H) return;
    int n = tid >> 6, c = tid & 63;
    const float* xr = x + (size_t)n * 5;
    float acc = 0.f;
    for (int k = 0; k < 5; ++k) acc += xr[k] * w[k * H + c];
    h0[tid] = acc;
}

__global__ void k_deg(const int* __restrict__ dst, float* __restrict__ deg, int E)
{
    int e = blockIdx.x * blockDim.x + threadIdx.x;
    if (e >= E) return;
    atomicAdd(deg + dst[e], 1.0f);
}

__global__ void k_dinv(const float* __restrict__ deg, float* __restrict__ dinv, int N)
{
    int n = blockIdx.x * blockDim.x + threadIdx.x;
    if (n >= N) return;
    dinv[n] = rsqrtf(deg[n] + 1.0f);                // +1 self loop; always > 0
}

__global__ void k_gcn_edge(const int* __restrict__ src, const int* __restrict__ dst,
                           const float* __restrict__ h0, const float* __restrict__ dinv,
                           float* __restrict__ agg, int E)
{
    int tid = blockIdx.x * blockDim.x + threadIdx.x;
    if (tid >= E * 16) return;
    int e = tid >> 4, q = (tid & 15) << 2;
    int s = src[e], d = dst[e];
    float norm = dinv[s] * dinv[d];
    float4 hv = *reinterpret_cast<const float4*>(h0 + (size_t)s * H + q);
    float* ap = agg + (size_t)d * H + q;
    atomicAdd(ap + 0, hv.x * norm);
    atomicAdd(ap + 1, hv.y * norm);
    atomicAdd(ap + 2, hv.z * norm);
    atomicAdd(ap + 3, hv.w * norm);
}

__global__ void k_gcn_final(const float* __restrict__ agg, const float* __restrict__ h0,
                            const float* __restrict__ dinv, const float* __restrict__ cb,
                            const float* __restrict__ bg, const float* __restrict__ bb,
                            float* __restrict__ hout, int N)
{
    int tid = blockIdx.x * blockDim.x + threadIdx.x;
    if (tid >= N * H) return;
    int n = tid >> 6, c = tid & 63;
    float di = dinv[n];
    float v = agg[tid] + h0[tid] * di * di + cb[c];
    v = v * (bg[c] * rsqrtf(1.0f + 1e-5f)) + bb[c];   // eval-mode BN
    hout[tid] = fmaxf(v, 0.0f);
}

// =====================================================================
// GEN layer: LayerNorm + PReLU, then per-dst softmax aggregation
// =====================================================================
__global__ void k_ln_prelu(const float* __restrict__ h, const float* __restrict__ g,
                           const float* __restrict__ b, const float* __restrict__ a,
                           float* __restrict__ u, int N)
{
    int n = blockIdx.x * blockDim.x + threadIdx.x;
    if (n >= N) return;
    const float* row = h + (size_t)n * H;
    float mu = 0.f;
    for (int c = 0; c < H; ++c) mu += row[c];
    mu *= (1.0f / H);
    float var = 0.f;
    for (int c = 0; c < H; ++c) { float d = row[c] - mu; var += d * d; }
    var *= (1.0f / H);
    float inv = rsqrtf(var + 1e-5f);
    float* ur = u + (size_t)n * H;
    for (int c = 0; c < H; ++c) {
        float v = (row[c] - mu) * inv * g[c] + b[c];
        ur[c] = (v >= 0.f) ? v : a[c] * v;
    }
}

__global__ void k_edge_max(const int* __restrict__ src, const int* __restrict__ dst,
                           const float* __restrict__ u, const float* __restrict__ gen_t,
                           int layer, unsigned* __restrict__ mkey, int E)
{
    int tid = blockIdx.x * blockDim.x + threadIdx.x;
    if (tid >= E * 16) return;
    int e = tid >> 4, q = (tid & 15) << 2;
    int s = src[e], d = dst[e];
    float t = gen_t[layer];
    float4 uv = *reinterpret_cast<const float4*>(u + (size_t)s * H + q);
    unsigned* mk = mkey + (size_t)d * H + q;
    atomicMax(mk + 0, fkey((fmaxf(uv.x, 0.f) + 1e-7f) * t));
    atomicMax(mk + 1, fkey((fmaxf(uv.y, 0.f) + 1e-7f) * t));
    atomicMax(mk + 2, fkey((fmaxf(uv.z, 0.f) + 1e-7f) * t));
    atomicMax(mk + 3, fkey((fmaxf(uv.w, 0.f) + 1e-7f) * t));
}

__global__ void k_edge_sum(const int* __restrict__ src, const int* __restrict__ dst,
                           const float* __restrict__ u, const float* __restrict__ gen_t,
                           int layer, const unsigned* __restrict__ mkey,
                           float* __restrict__ sbuf, float* __restrict__ wbuf, int E)
{
    int tid = blockIdx.x * blockDim.x + threadIdx.x;
    if (tid >= E * 16) return;
    int e = tid >> 4, q = (tid & 15) << 2;
    int s = src[e], d = dst[e];
    float t = gen_t[layer];
    float4 uv = *reinterpret_cast<const float4*>(u + (size_t)s * H + q);
    size_t base = (size_t)d * H + q;
    const unsigned* mk = mkey + base;
    float* sp = sbuf + base;
    float* wp = wbuf + base;
    float msg, ex;
    msg = fmaxf(uv.x, 0.f) + 1e-7f; ex = __expf(msg * t - funkey(mk[0]));
    atomicAdd(sp + 0, ex); atomicAdd(wp + 0, msg * ex);
    msg = fmaxf(uv.y, 0.f) + 1e-7f; ex = __expf(msg * t - funkey(mk[1]));
    atomicAdd(sp + 1, ex); atomicAdd(wp + 1, msg * ex);
    msg = fmaxf(uv.z, 0.f) + 1e-7f; ex = __expf(msg * t - funkey(mk[2]));
    atomicAdd(sp + 2, ex); atomicAdd(wp + 2, msg * ex);
    msg = fmaxf(uv.w, 0.f) + 1e-7f; ex = __expf(msg * t - funkey(mk[3]));
    atomicAdd(sp + 3, ex); atomicAdd(wp + 3, msg * ex);
}

// agg = (s>0 ? w/s : 0) + u, written in-place into w
__global__ void k_agg(const float* __restrict__ sbuf, float* __restrict__ wbuf,
                      const float* __restrict__ u, int N)
{
    int tid = blockIdx.x * blockDim.x + threadIdx.x;
    if (tid >= N * H) return;
    float sv = sbuf[tid];
    float acc = (sv > 0.f) ? (wbuf[tid] / sv) : 0.f;
    wbuf[tid] = acc + u[tid];
}

// =====================================================================
// JumpingKnowledge pooling (per 64-channel block) + readout
// =====================================================================
__global__ void k_pool(const float* __restrict__ h, int colOff, const int* __restrict__ batch,
                       float* __restrict__ psum, unsigned* __restrict__ pmax,
                       float* __restrict__ cnt, int N, int doCnt)
{
    int tid = blockIdx.x * blockDim.x + threadIdx.x;
    if (tid >= N * 16) return;
    int n = tid >> 4, q = (tid & 15) << 2;
    int g = batch[n];
    float4 v = *reinterpret_cast<const float4*>(h + (size_t)n * H + q);
    size_t base = (size_t)g * 256 + colOff + q;
    atomicAdd(psum + base + 0, v.x); atomicMax(pmax + base + 0, fkey(v.x));
    atomicAdd(psum + base + 1, v.y); atomicMax(pmax + base + 1, fkey(v.y));
    atomicAdd(psum + base + 2, v.z); atomicMax(pmax + base + 2, fkey(v.z));
    atomicAdd(psum + base + 3, v.w); atomicMax(pmax + base + 3, fkey(v.w));
    if (doCnt && (tid & 15) == 0) atomicAdd(cnt + g, 1.0f);
}

__global__ void k_passemble(const float* __restrict__ psum, const unsigned* __restrict__ pmax,
                            const float* __restrict__ cnt, float* __restrict__ p, int G)
{
    int tid = blockIdx.x * blockDim.x + threadIdx.x;
    if (tid >= G * 512) return;
    int g = tid >> 9, c = tid & 511;
    float v;
    if (c < 256) {
        v = psum[(size_t)g * 256 + c] / fmaxf(cnt[g], 1.0f);
    } else {
        float m = funkey(pmax[(size_t)g * 256 + (c - 256)]);
        v = (cnt[g] > 0.f) ? m : 0.f;     // empty segment -> 0 (isfinite guard)
    }
    p[(size_t)g * 512 + c] = v;
}

__global__ void k_out(const float* __restrict__ r2, const float* __restrict__ ow,
                      const float* __restrict__ ob, float* __restrict__ out, int G)
{
    int g = blockIdx.x * blockDim.x + threadIdx.x;
    if (g >= G) return;
    const float* r = r2 + (size_t)g * H;
    float acc = 0.f;
    for (int c = 0; c < H; ++c) acc += r[c] * ow[c];
    out[g] = acc + ob[0];
}

// =====================================================================
// Host launcher
// =====================================================================
extern "C" void kernel_launch(void* const* d_in, const int* in_sizes, int n_in,
                              void* d_out, int out_size, void* d_ws, size_t ws_size,
                              hipStream_t stream)
{
    (void)n_in; (void)ws_size;
    const float* x        = (const float*)d_in[0];
    const int*   eidx     = (const int*)  d_in[1];
    const int*   batch    = (const int*)  d_in[2];
    const float* conv1_W  = (const float*)d_in[3];
    const float* conv1_b  = (const float*)d_in[4];
    const float* bn1_g    = (const float*)d_in[5];
    const float* bn1_b    = (const float*)d_in[6];
    const float* ln_g     = (const float*)d_in[7];
    const float* ln_b     = (const float*)d_in[8];
    const float* prelu_a  = (const float*)d_in[9];
    const float* gen_t    = (const float*)d_in[10];
    const float* mlp_W1   = (const float*)d_in[11];
    const float* mlp_b1   = (const float*)d_in[12];
    const float* mlp_bn_g = (const float*)d_in[13];
    const float* mlp_bn_b = (const float*)d_in[14];
    const float* mlp_W2   = (const float*)d_in[15];
    const float* mlp_b2   = (const float*)d_in[16];
    const float* lin1_W   = (const float*)d_in[17];
    const float* lin1_b   = (const float*)d_in[18];
    const float* lin2_W   = (const float*)d_in[19];
    const float* lin2_b   = (const float*)d_in[20];
    const float* out_W    = (const float*)d_in[21];
    const float* out_b    = (const float*)d_in[22];

    const int N = in_sizes[2];          // batch_idx has N entries
    const int E = in_sizes[1] / 2;      // edge_index is (2,E)
    const int G = out_size;             // output is (G,1)
    const int* src = eidx;
    const int* dst = eidx + E;

    // ---- workspace carve-out (~79 MB; everything mutable is re-inited each call) ----
    size_t cur = 0;
    auto alloc = [&](size_t bytes) -> char* {
        cur = (cur + 255) & ~(size_t)255;
        char* p = (char*)d_ws + cur;
        cur += bytes;
        return p;
    };
    const size_t NH = (size_t)N * H * sizeof(float);
    float*    hA   = (float*)alloc(NH);
    float*    hB   = (float*)alloc(NH);
    float*    u    = (float*)alloc(NH);       // also GCN aggregation buffer
    char*     trio = alloc(3 * NH);           // [mkey | s | w] contiguous
    unsigned* mkey = (unsigned*)trio;
    float*    sbuf = (float*)(trio + NH);
    float*    wbuf = (float*)(trio + 2 * NH);
    float*    z1   = (float*)trio;            // N x 2H, aliases mkey+s (dead by then)
    float*    h0pre= sbuf;                    // aliases s (dead before layer 0 memset)
    float*    deg  = (float*)alloc((size_t)N * sizeof(float));
    float*    dinv = (float*)alloc((size_t)N * sizeof(float));
    char*     poolc= alloc(((size_t)2 * G * 256 + G) * sizeof(float));
    float*    psum = (float*)poolc;
    unsigned* pmax = (unsigned*)(poolc + (size_t)G * 256 * sizeof(float));
    float*    cnt  = (float*)(poolc + (size_t)2 * G * 256 * sizeof(float));
    float*    p    = (float*)alloc((size_t)G * 512 * sizeof(float));
    float*    r1   = (float*)alloc((size_t)G * 128 * sizeof(float));
    float*    r2   = (float*)alloc((size_t)G * 64 * sizeof(float));

    const int TB = 256;
    const int gNH  = (N * H + TB - 1) / TB;
    const int gE16 = (E * 16 + TB - 1) / TB;
    const int gN16 = (N * 16 + TB - 1) / TB;
    const int gN   = (N + TB - 1) / TB;
    const int gE   = (E + TB - 1) / TB;
    const float BNS = 1.0f / sqrtf(1.0f + 1e-5f);

    // ---- GCNConv ----
    hipMemsetAsync(deg, 0, (size_t)N * sizeof(float), stream);
    hipMemsetAsync(u, 0, NH, stream);                        // zero GCN agg
    hipMemsetAsync(poolc, 0, ((size_t)2 * G * 256 + G) * sizeof(float), stream);
    k_h0pre<<<gNH, TB, 0, stream>>>(x, conv1_W, h0pre, N);
    k_deg<<<gE, TB, 0, stream>>>(dst, deg, E);
    k_dinv<<<gN, TB, 0, stream>>>(deg, dinv, N);
    k_gcn_edge<<<gE16, TB, 0, stream>>>(src, dst, h0pre, dinv, u, E);
    k_gcn_final<<<gNH, TB, 0, stream>>>(u, h0pre, dinv, conv1_b, bn1_g, bn1_b, hA, N);
    k_pool<<<gN16, TB, 0, stream>>>(hA, 0, batch, psum, pmax, cnt, N, 1);

    // ---- DeepGCN layers ----
    float* curh = hA;
    float* nxth = hB;
    for (int i = 0; i < NLAYERS; ++i) {
        hipMemsetAsync(trio, 0, 3 * NH, stream);             // mkey(=key0 identity), s, w
        k_ln_prelu<<<gN, TB, 0, stream>>>(curh, ln_g + i * H, ln_b + i * H,
                                          prelu_a + i * H, u, N);
        k_edge_max<<<gE16, TB, 0, stream>>>(src, dst, u, gen_t, i, mkey, E);
        k_edge_sum<<<gE16, TB, 0, stream>>>(src, dst, u, gen_t, i, mkey, sbuf, wbuf, E);
        k_agg<<<gNH, TB, 0, stream>>>(sbuf, wbuf, u, N);     // agg -> wbuf

        // z1 = relu(bn(agg @ W1 + b1))  [N x 128]
        dim3 g1((N + 15) / 16, (2 * H) / 16);
        k_wmma_gemm<<<g1, 32, 0, stream>>>(
            wbuf, H, mlp_W1 + (size_t)i * H * 2 * H, 2 * H,
            mlp_b1 + i * 2 * H, mlp_bn_g + i * 2 * H, BNS, mlp_bn_b + i * 2 * H,
            nullptr, 0, z1, 2 * H, N, H, 1);

        // h_next = h + (z1 @ W2 + b2)  [N x 64]
        dim3 g2((N + 15) / 16, H / 16);
        k_wmma_gemm<<<g2, 32, 0, stream>>>(
            z1, 2 * H, mlp_W2 + (size_t)i * 2 * H * H, H,
            mlp_b2 + i * H, nullptr, 1.0f, nullptr,
            curh, H, nxth, H, N, 2 * H, 0);

        k_pool<<<gN16, TB, 0, stream>>>(nxth, (i + 1) * H, batch, psum, pmax, cnt, N, 0);
        float* t = curh; curh = nxth; nxth = t;
    }

    // ---- readout ----
    k_passemble<<<(G * 512 + TB - 1) / TB, TB, 0, stream>>>(psum, pmax, cnt, p, G);
    dim3 gl1((G + 15) / 16, 128 / 16);
    k_wmma_gemm<<<gl1, 32, 0, stream>>>(p, 512, lin1_W, 128, lin1_b, nullptr, 1.0f,
                                        nullptr, nullptr, 0, r1, 128, G, 512, 1);
    dim3 gl2((G + 15) / 16, 64 / 16);
    k_wmma_gemm<<<gl2, 32, 0, stream>>>(r1, 128, lin2_W, 64, lin2_b, nullptr, 1.0f,
                                        nullptr, nullptr, 0, r2, 64, G, 128, 1);
    k_out<<<(G + TB - 1) / TB, TB, 0, stream>>>(r2, out_W, out_b, (float*)d_out, G);
}